// Voxelizer_62826781606551
// MI455X (gfx1250) — compile-verified
//
#include <hip/hip_runtime.h>
#include <hip/hip_bf16.h>
#include <math.h>

// ---------------------------------------------------------------------------
// Voxelizer: x (16, 262144, 3) f32 -> dists (16, 4096, 12) f32
//   per batch: min/max -> voxel size; scatter points into 16^3 voxels;
//   accumulate count, sum(3), sum of outer products (6 unique, symmetric);
//   finalize mean + covariance (mean⊗mean via V_WMMA_F32_16X16X4_F32),
//   zero where count <= 1.
// ---------------------------------------------------------------------------

#define GRID_G   16
#define NVOX     4096              // 16^3
#define NBATCH   16
#define NPTS     262144            // points per batch
#define EPSF     1e-6f

#define ACC_FLOATS (NVOX * 10)     // 40960 floats
#define ACC_BYTES  (ACC_FLOATS * 4)        // 163840 B of LDS accumulators
#define TILE_PTS   1024
#define TILE_BYTES (TILE_PTS * 12)         // 12288 B per staged tile
#define SMEM_BYTES (ACC_BYTES + 2 * TILE_BYTES)  // 188416 B (< 320KB WGP LDS)

#define CHUNKS_PER_BATCH 4
#define CHUNK_PTS  (NPTS / CHUNKS_PER_BATCH)     // 65536
#define TILES_PER_CHUNK (CHUNK_PTS / TILE_PTS)   // 64

// workspace layout (as uint32/float words):
//   [0  .. 47]  : encoded per-(batch,axis) min keys   (16*3)
//   [48 .. 95]  : encoded per-(batch,axis) max keys
//   [96 .. 127] : pad
//   [128 .. 128 + 16*4096*10) : float accumulators [b][v][10]
#define WS_KEYS_MIN 0
#define WS_KEYS_MAX 48
#define WS_ACC_OFF  128
#define WS_WORDS    (WS_ACC_OFF + NBATCH * ACC_FLOATS)

typedef __attribute__((ext_vector_type(2))) float v2f;
typedef __attribute__((ext_vector_type(8))) float v8f;

// ---- order-preserving float <-> uint encoding for atomic min/max ----------
__device__ __forceinline__ unsigned f2key(float f) {
  unsigned u = __float_as_uint(f);
  return (u & 0x80000000u) ? ~u : (u | 0x80000000u);
}
__device__ __forceinline__ float key2f(unsigned k) {
  unsigned u = (k & 0x80000000u) ? (k ^ 0x80000000u) : ~k;
  return __uint_as_float(u);
}

// ---- CDNA5 async global -> LDS (tracked by ASYNCcnt) ----------------------
__device__ __forceinline__ void async_b128_to_lds(unsigned lds_byte_addr,
                                                  const void* gaddr) {
  asm volatile("global_load_async_to_lds_b128 %0, %1, off"
               :
               : "v"(lds_byte_addr), "v"(gaddr)
               : "memory");
}
__device__ __forceinline__ void wait_async_le(int n) {
  if (n == 0) asm volatile("s_wait_asynccnt 0x0" ::: "memory");
  else        asm volatile("s_wait_asynccnt 0x1" ::: "memory");
}

__device__ __forceinline__ void atomAddF(float* p, float v) {
  unsafeAtomicAdd(p, v);   // ds_add_f32 (LDS) / global_atomic_add_f32 (global)
}

// ---------------------------------------------------------------------------
// K0: initialize workspace (must run every launch: determinism requirement)
// ---------------------------------------------------------------------------
__global__ void vox_init_ws(unsigned* ws) {
  int stride = gridDim.x * blockDim.x;
  for (int i = blockIdx.x * blockDim.x + threadIdx.x; i < WS_WORDS; i += stride) {
    unsigned v = 0u;
    if (i < WS_KEYS_MAX) v = 0xFFFFFFFFu;  // min sentinel = +inf key
    ws[i] = v;                              // max sentinel / accum zeros = 0
  }
}

// ---------------------------------------------------------------------------
// K1: per-batch min/max over N points, via encoded-uint atomics
//   grid = NBATCH * 32 blocks of 512 threads; 8192 points per block
// ---------------------------------------------------------------------------
__global__ __launch_bounds__(512) void vox_minmax(const float* __restrict__ x,
                                                  unsigned* __restrict__ ws) {
  __shared__ unsigned smin[3], smax[3];
  const int tid   = threadIdx.x;
  const int b     = blockIdx.x >> 5;
  const int slice = blockIdx.x & 31;
  if (tid < 3) { smin[tid] = 0xFFFFFFFFu; smax[tid] = 0u; }
  __syncthreads();

  const float* xb = x + (size_t)b * NPTS * 3;
  const size_t p0 = (size_t)slice * 8192;

  float lmn[3] = { 3.4e38f,  3.4e38f,  3.4e38f};
  float lmx[3] = {-3.4e38f, -3.4e38f, -3.4e38f};
  for (int i = tid; i < 8192; i += 512) {
    const float* p = xb + (p0 + i) * 3;
    #pragma unroll
    for (int a = 0; a < 3; ++a) {
      float v = p[a];
      lmn[a] = fminf(lmn[a], v);
      lmx[a] = fmaxf(lmx[a], v);
    }
  }
  #pragma unroll
  for (int a = 0; a < 3; ++a) {
    atomicMin(&smin[a], f2key(lmn[a]));   // ds_min_u32
    atomicMax(&smax[a], f2key(lmx[a]));   // ds_max_u32
  }
  __syncthreads();
  if (tid < 3) {
    atomicMin(&ws[WS_KEYS_MIN + b * 3 + tid], smin[tid]);
    atomicMax(&ws[WS_KEYS_MAX + b * 3 + tid], smax[tid]);
  }
}

// ---------------------------------------------------------------------------
// K2: scatter-accumulate. One block = one quarter of one batch (65536 pts).
//   160KB LDS voxel accumulator + double-buffered async-staged point tiles.
// ---------------------------------------------------------------------------
__global__ __launch_bounds__(1024) void vox_accumulate(const float* __restrict__ x,
                                                       float* __restrict__ gacc_all,
                                                       const unsigned* __restrict__ ws) {
  extern __shared__ char smem[];
  float* acc = (float*)smem;                    // [NVOX * 10]
  const int tid   = threadIdx.x;
  const int b     = blockIdx.x >> 2;            // CHUNKS_PER_BATCH == 4
  const int chunk = blockIdx.x & 3;

  // zero the LDS voxel accumulator
  for (int i = tid; i < ACC_FLOATS; i += 1024) acc[i] = 0.0f;

  // per-batch min & voxel size (uniform; scalar loads)
  float mn[3], vs[3];
  #pragma unroll
  for (int a = 0; a < 3; ++a) {
    float lo = key2f(ws[WS_KEYS_MIN + b * 3 + a]);
    float hi = key2f(ws[WS_KEYS_MAX + b * 3 + a]);
    mn[a] = lo;
    vs[a] = (hi - lo) * (1.0f / GRID_G) + EPSF;
  }

  const char* gbase =
      (const char*)(x + ((size_t)b * NPTS + (size_t)chunk * CHUNK_PTS) * 3);
  // low 32 bits of a flat LDS address == wave-relative LDS byte offset
  const unsigned lds_base = (unsigned)(size_t)smem;
  const unsigned stage0   = lds_base + ACC_BYTES;
  const bool loader = (tid < TILE_BYTES / 16);  // 768 lanes = waves 0..23

  __syncthreads();

  // prefetch tile 0
  if (loader)
    async_b128_to_lds(stage0 + (unsigned)tid * 16u, gbase + (size_t)tid * 16);

  for (int t = 0; t < TILES_PER_CHUNK; ++t) {
    const int buf  = t & 1;
    const int nbuf = buf ^ 1;
    const bool more = (t + 1 < TILES_PER_CHUNK);
    if (more && loader)
      async_b128_to_lds(stage0 + (unsigned)(nbuf * TILE_BYTES) + (unsigned)tid * 16u,
                        gbase + (size_t)(t + 1) * TILE_BYTES + (size_t)tid * 16);
    wait_async_le(more ? 1 : 0);   // tile t landed; tile t+1 may stay in flight
    __syncthreads();

    const float* pts = (const float*)(smem + ACC_BYTES + buf * TILE_BYTES);
    const float px = pts[tid * 3 + 0];
    const float py = pts[tid * 3 + 1];
    const float pz = pts[tid * 3 + 2];

    int i0 = (int)floorf((px - mn[0]) / vs[0]);
    int i1 = (int)floorf((py - mn[1]) / vs[1]);
    int i2 = (int)floorf((pz - mn[2]) / vs[2]);
    i0 = min(GRID_G - 1, max(0, i0));
    i1 = min(GRID_G - 1, max(0, i1));
    i2 = min(GRID_G - 1, max(0, i2));
    const int v = i0 * (GRID_G * GRID_G) + i1 * GRID_G + i2;

    float* a = acc + v * 10;
    atomAddF(a + 0, 1.0f);
    atomAddF(a + 1, px);
    atomAddF(a + 2, py);
    atomAddF(a + 3, pz);
    atomAddF(a + 4, px * px);
    atomAddF(a + 5, px * py);
    atomAddF(a + 6, px * pz);
    atomAddF(a + 7, py * py);
    atomAddF(a + 8, py * pz);
    atomAddF(a + 9, pz * pz);

    __syncthreads();
  }

  // flush LDS accumulator into the per-batch global accumulator
  float* gacc = gacc_all + (size_t)b * ACC_FLOATS;
  for (int i = tid; i < ACC_FLOATS; i += 1024) {
    float val = acc[i];
    if (val != 0.0f) atomAddF(gacc + i, val);
  }
}

// ---------------------------------------------------------------------------
// K3: finalize mean + covariance. One wave finalizes 4 voxels; the mean⊗mean
//   outer products for all 4 voxels are produced by ONE v_wmma_f32_16x16x4_f32
//   on a block-diagonal operand:
//     A[r][k] = (k == r/4) ? mean[r/4][r%4] : 0   (r%4==3 row is zero pad)
//     B[k][c] = (k == c/4) ? mean[c/4][c%4] : 0
//   => D[r][c] = (r/4 == c/4) ? mean[v][r%4]*mean[v][c%4] : 0
//   Per the CDNA5 VGPR layouts (A: lane holds M=lane%16, K split {0,1}/{2,3}
//   across lane halves; B: lane holds N=lane%16, same K split) the per-lane
//   register contents of A and B are identical, so one v2f feeds both.
// ---------------------------------------------------------------------------
__global__ __launch_bounds__(256) void vox_finalize_wmma(const float* __restrict__ gacc,
                                                         float* __restrict__ out) {
  const int tid   = threadIdx.x;
  const int lane  = tid & 31;
  const int wave  = blockIdx.x * (256 / 32) + (tid >> 5);
  const int vbase = wave * 4;                 // 4 voxels per wave
  const int r     = lane & 15;                // A's M index == B's N index
  const int vsel  = r >> 2;                   // voxel slot 0..3 in this wave
  const int comp  = r & 3;                    // component 0..3 (3 = zero pad)
  const bool hi   = lane >= 16;
  const int k0    = hi ? 2 : 0;               // K indices held by this lane
  const int k1    = k0 + 1;

  const float* av = gacc + (size_t)(vbase + vsel) * 10;
  const float n   = av[0];
  const float sn  = fmaxf(n, 1.0f);
  const float m   = (comp < 3) ? (av[1 + comp] / sn) : 0.0f;   // mean component

  v2f ab;
  ab.x = (k0 == vsel) ? m : 0.0f;
  ab.y = (k1 == vsel) ? m : 0.0f;
  v8f cz = {};
  // D = A x B (+0): 4 voxels' 3x3 mean outer products in one matrix op.
  // No divergence above: EXEC is all-ones here (ISA requirement).
  v8f d = __builtin_amdgcn_wmma_f32_16x16x4_f32(
      /*neg_a=*/false, ab, /*neg_b=*/false, ab,
      /*c_mod=*/(short)0, cz, /*reuse_a=*/false, /*reuse_b=*/false);

  const float dn    = fmaxf(n - 1.0f, 1.0f);
  const bool  valid = (n > 1.0f);             // COUNT_THRES = 1
  float* o = out + (size_t)(vbase + vsel) * 12;

  if (comp < 3) {
    // mean: written once (low half lanes), zeroed when invalid
    if (!hi) o[comp] = valid ? m : 0.0f;
    // covariance column b=comp of voxel vsel: this lane's half of D holds
    // M = g (low half) / g+8 (high half); we own rows M = 4*vsel + a, which
    // fold to register index g = (vsel&1)*4 + a within the owning half.
    const bool ownHalf = hi ? (vsel >= 2) : (vsel < 2);
    if (ownHalf) {
      const bool odd = (vsel & 1) != 0;
      #pragma unroll
      for (int a = 0; a < 3; ++a) {
        // symmetric-storage index of sq[a][comp], branch-free arithmetic:
        //   a=0 -> 4+comp ; a=1 -> {5,7,8}[comp] ; a=2 -> {6,8,9}[comp]
        int symIdx;
        if (a == 0)      symIdx = 4 + comp;
        else if (a == 1) symIdx = (comp == 0) ? 5 : (6 + comp);     // 5,7,8
        else             symIdx = (comp == 0) ? 6 : (7 + comp);     // 6,8,9
        const float sq = av[symIdx];
        const float mm = odd ? d[4 + a] : d[a];   // two-way register select
        o[3 + a * 3 + comp] = valid ? ((sq - n * mm) / dn) : 0.0f;
      }
    }
  }
}

// ---------------------------------------------------------------------------
extern "C" void kernel_launch(void* const* d_in, const int* in_sizes, int n_in,
                              void* d_out, int out_size, void* d_ws, size_t ws_size,
                              hipStream_t stream) {
  const float* x  = (const float*)d_in[0];
  float* out      = (float*)d_out;
  unsigned* ws    = (unsigned*)d_ws;
  float* gacc     = (float*)d_ws + WS_ACC_OFF;

  vox_init_ws<<<256, 256, 0, stream>>>(ws);
  vox_minmax<<<NBATCH * 32, 512, 0, stream>>>(x, ws);
  vox_accumulate<<<NBATCH * CHUNKS_PER_BATCH, 1024, SMEM_BYTES, stream>>>(x, gacc, ws);
  // 65536 voxels / (8 waves * 4 voxels per block) = 2048 blocks, all waves full
  vox_finalize_wmma<<<2048, 256, 0, stream>>>(gacc, out);
}